// MIRNN_91199335563546
// MI455X (gfx1250) — compile-verified
//
#include <hip/hip_runtime.h>
#include <math.h>

#define TSTEPS 32768
#define FDIM   256
#define HDIM   1024

// scan kernel geometry: 32 persistent workgroups, 32 rows (= 32 waves) each
#define NWG    32
#define ROWS   32
#define TPB    1024
#define SMEM_BYTES ((ROWS * HDIM + HDIM) * (int)sizeof(float))

typedef __attribute__((ext_vector_type(2))) float v2f;
typedef __attribute__((ext_vector_type(8))) float v8f;

// --------------------------------------------------------------------------
// Kernel A: xW[T,H] = x[T,F] @ Wih[H,F]^T using V_WMMA_F32_16X16X4_F32.
// One wave computes one 16x16 output tile; 8 waves/block share a t-tile.
// A-layout (16x4): lane m=lane&15 holds K = 2*(lane>>4) + {0,1}
// B-layout (4x16): lane n=lane&15 holds rows K = 2*(lane>>4) + {0,1}
// C-layout: VGPR r -> (M=r, N=lane) low lanes, (M=r+8, N=lane-16) high lanes
// --------------------------------------------------------------------------
__global__ __launch_bounds__(256) void mirnn_gemm_xw(
    const float* __restrict__ x,    // [T, F]
    const float* __restrict__ wih,  // [H, F]
    float* __restrict__ xw)         // [T, H]
{
    const int lane = threadIdx.x & 31;
    const int wave = threadIdx.x >> 5;
    const int t0 = blockIdx.y * 16;
    const int h0 = (blockIdx.x * 8 + wave) * 16;
    const int m   = lane & 15;
    const int khi = lane >> 4;   // 0 or 1

    const float* arow = x   + (size_t)(t0 + m) * FDIM + 2 * khi;
    const float* brow = wih + (size_t)(h0 + m) * FDIM + 2 * khi;

    v8f c = {};
#pragma unroll 8
    for (int k0 = 0; k0 < FDIM; k0 += 4) {
        v2f a = *(const v2f*)(arow + k0);
        v2f b = *(const v2f*)(brow + k0);
        c = __builtin_amdgcn_wmma_f32_16x16x4_f32(
                /*neg_a=*/false, a, /*neg_b=*/false, b,
                /*c_mod=*/(short)0, c, /*reuse_a=*/false, /*reuse_b=*/false);
    }

    float* outb = xw + (size_t)t0 * HDIM + h0 + m;
#pragma unroll
    for (int r = 0; r < 8; ++r)
        outb[(size_t)(r + 8 * khi) * HDIM] = c[r];
}

// --------------------------------------------------------------------------
// Init: zero the h ping-pong buffers and the global barrier counter.
// --------------------------------------------------------------------------
__global__ void mirnn_init(float* hbuf, unsigned* ctr) {
    int i = blockIdx.x * blockDim.x + threadIdx.x;
    if (i < 2 * HDIM) hbuf[i] = 0.0f;
    if (i == 0) *ctr = 0u;
}

// --------------------------------------------------------------------------
// Kernel B: persistent recurrent scan. Each WG keeps a 32-row slice of W_hh
// in LDS; each wave owns one hidden row for the whole run. Device-wide
// barrier via monotonic atomic counter (release fence + agent-scope spin).
// --------------------------------------------------------------------------
__global__ __launch_bounds__(TPB) void mirnn_scan(
    const float* __restrict__ xw,    // [T, H]
    const float* __restrict__ whh,   // [H, H]
    const float* __restrict__ wvec,  // [H]
    const float* __restrict__ bih,   // [H]
    const float* __restrict__ bhh,   // [H]
    const float* __restrict__ bvec,  // [H]
    float* __restrict__ out,         // [T, H]
    float* __restrict__ hbuf,        // [2, H] ping-pong
    unsigned* __restrict__ ctr)
{
    extern __shared__ float smem[];
    float* wlds = smem;               // ROWS * HDIM
    float* hlds = smem + ROWS * HDIM; // HDIM

    const int tid  = threadIdx.x;
    const int lane = tid & 31;
    const int wave = tid >> 5;                 // local row 0..31
    const int row  = blockIdx.x * ROWS + wave; // global hidden row

    // stage this WG's weight slice into LDS
    {
        const size_t base = (size_t)blockIdx.x * ROWS * HDIM;
        for (int i = tid; i < ROWS * HDIM; i += TPB)
            wlds[i] = whh[base + i];
    }
    // per-row constants live in registers for all 32768 steps
    const float wr   = wvec[row];
    const float bihr = bih[row];
    const float bhhr = bhh[row];
    const float br   = bvec[row];
    __syncthreads();

    const float* wrow = wlds + wave * HDIM;

    for (int t = 0; t < TSTEPS; ++t) {
        const float* hcur = hbuf + ((t) & 1) * HDIM;
        float*       hnxt = hbuf + ((t + 1) & 1) * HDIM;

        // stage h(t-1) into LDS with device-coherent loads (skip stale L0)
        hlds[tid] = __hip_atomic_load(hcur + tid, __ATOMIC_RELAXED,
                                      __HIP_MEMORY_SCOPE_AGENT);
        float xwv = 0.0f;
        if (lane == 0) {
            xwv = xw[(size_t)t * HDIM + row];
            if (t + 16 < TSTEPS)
                __builtin_prefetch(xw + (size_t)(t + 16) * HDIM + row, 0, 0);
        }
        __syncthreads();

        // 32-element bank-swizzled partial dot per lane
        float acc = 0.0f;
        const int cbase = lane * 32;
#pragma unroll 8
        for (int j = 0; j < 32; ++j) {
            const int cc = cbase + ((j + lane) & 31);
            acc = fmaf(wrow[cc], hlds[cc], acc);
        }
        // wave32 all-reduce
#pragma unroll
        for (int off = 16; off > 0; off >>= 1)
            acc += __shfl_xor(acc, off, 32);

        if (lane == 0) {
            const float hw = acc;
            const float v = tanhf(xwv * (wr * hw + bihr) + bhhr * hw + br);
            out[(size_t)t * HDIM + row] = v;
            __hip_atomic_store(hnxt + row, v, __ATOMIC_RELAXED,
                               __HIP_MEMORY_SCOPE_AGENT);
            // this lane's stores must reach device scope before the WG signals
            __builtin_amdgcn_fence(__ATOMIC_RELEASE, "agent");
        }
        __syncthreads();

        // device-wide barrier: monotonic counter
        if (tid == 0) {
            __hip_atomic_fetch_add(ctr, 1u, __ATOMIC_RELAXED,
                                   __HIP_MEMORY_SCOPE_AGENT);
            const unsigned target = (unsigned)NWG * (unsigned)(t + 1);
            while (__hip_atomic_load(ctr, __ATOMIC_RELAXED,
                                     __HIP_MEMORY_SCOPE_AGENT) < target)
                __builtin_amdgcn_s_sleep(1);
        }
        __syncthreads();
    }
}

// --------------------------------------------------------------------------
extern "C" void kernel_launch(void* const* d_in, const int* in_sizes, int n_in,
                              void* d_out, int out_size, void* d_ws, size_t ws_size,
                              hipStream_t stream) {
    const float* x    = (const float*)d_in[0]; // [T,1,F]
    const float* wih  = (const float*)d_in[1]; // [H,F]
    const float* whh  = (const float*)d_in[2]; // [H,H]
    const float* wvec = (const float*)d_in[3]; // [H]
    const float* bihv = (const float*)d_in[4]; // [H]
    const float* bhhv = (const float*)d_in[5]; // [H]
    const float* bvec = (const float*)d_in[6]; // [H]
    float* out = (float*)d_out;                // [T,1,H]

    float*    xw   = (float*)d_ws;                       // T*H floats
    float*    hbuf = xw + (size_t)TSTEPS * HDIM;         // 2*H floats
    unsigned* ctr  = (unsigned*)(hbuf + 2 * HDIM);       // 1 u32

    (void)ws_size; (void)in_sizes; (void)n_in; (void)out_size;

    hipFuncSetAttribute(reinterpret_cast<const void*>(mirnn_scan),
                        hipFuncAttributeMaxDynamicSharedMemorySize, SMEM_BYTES);

    mirnn_init<<<dim3((2 * HDIM + 255) / 256), dim3(256), 0, stream>>>(hbuf, ctr);
    mirnn_gemm_xw<<<dim3(HDIM / 128, TSTEPS / 16), dim3(256), 0, stream>>>(x, wih, xw);
    mirnn_scan<<<dim3(NWG), dim3(TPB), SMEM_BYTES, stream>>>(
        xw, whh, wvec, bihv, bhhv, bvec, out, hbuf, ctr);
}